// ECELoss_62758062129747
// MI455X (gfx1250) — compile-verified
//
#include <hip/hip_runtime.h>

typedef __attribute__((ext_vector_type(2))) float v2f;
typedef __attribute__((ext_vector_type(8))) float v8f;

#define NBINS      15
#define CDIM       32
#define TPB        256
#define MAX_BLOCKS 512
#define NCOPIES    16
#define CSTRIDE    49   // odd dword stride -> 16 copies land on 16 distinct LDS banks
#define PARTIAL_F  48   // per-block partial: [3 quantities][16 bins] floats

// ---------------------------------------------------------------------------
// Pass 1: bandwidth-bound streaming pass. Each thread owns whole rows
// (8 x b128 loads), computes max/argmax/bin, accumulates into a bank-skewed
// LDS histogram (ds_add_f32, <=2-way same-address conflict), then the block
// writes a 48-float partial [count|accSum|confSum][16 bins] to workspace.
// ---------------------------------------------------------------------------
__global__ __launch_bounds__(TPB)
void ece_partial_kernel(const float* __restrict__ soft,
                        const int* __restrict__ labels,
                        float* __restrict__ partials,
                        int n, int nblocks) {
  __shared__ float hist[NCOPIES * CSTRIDE];
  const int tid = threadIdx.x;
  for (int i = tid; i < NCOPIES * CSTRIDE; i += TPB) hist[i] = 0.f;
  __syncthreads();

  float* myh = &hist[(tid & (NCOPIES - 1)) * CSTRIDE];
  const int stride = nblocks * TPB;

  for (int i = blockIdx.x * TPB + tid; i < n; i += stride) {
    const float4* row = (const float4*)(soft + (size_t)i * CDIM);
    size_t pi = (size_t)i + (size_t)stride;
    if (pi < (size_t)n)
      __builtin_prefetch(soft + pi * CDIM, 0, 0);   // global_prefetch_b8

    float best = -1.f;
    int   barg = 0;
#pragma unroll
    for (int v = 0; v < 8; ++v) {
      float4 q = row[v];
      int j = v * 4;
      if (q.x > best) { best = q.x; barg = j;     }
      if (q.y > best) { best = q.y; barg = j + 1; }
      if (q.z > best) { best = q.z; barg = j + 2; }
      if (q.w > best) { best = q.w; barg = j + 3; }
    }
    float acc = (barg == labels[i]) ? 1.f : 0.f;
    // bin i covers (i/15, (i+1)/15]  ->  ceil(c*15)-1, clamped
    int b = (int)ceilf(best * (float)NBINS) - 1;
    b = b < 0 ? 0 : (b > NBINS - 1 ? NBINS - 1 : b);

    atomicAdd(&myh[0 * 16 + b], 1.f);    // count
    atomicAdd(&myh[1 * 16 + b], acc);    // accuracy sum
    atomicAdd(&myh[2 * 16 + b], best);   // confidence sum
  }
  __syncthreads();

  if (tid < PARTIAL_F) {
    float s = 0.f;
#pragma unroll
    for (int c = 0; c < NCOPIES; ++c) s += hist[c * CSTRIDE + tid];
    partials[(size_t)blockIdx.x * PARTIAL_F + tid] = s;
  }
}

// ---------------------------------------------------------------------------
// Pass 2: single wave32. Reduce nblocks x [3][16] partials with
// V_WMMA_F32_16X16X4_F32:  D += ones(16x4) x B(4x16), where B holds the
// 16-bin partial of 4 blocks (one per K row). A is constant and the K sum is
// symmetric, so the result is independent of the exact B fragment layout.
// D[0][n] (lane n, c[0]) = total over all blocks for bin n.
// ---------------------------------------------------------------------------
__global__ __launch_bounds__(32)
void ece_final_kernel(const float* __restrict__ partials,
                      float* __restrict__ out,
                      int nblocks, int n) {
  const int lane = threadIdx.x;
  const int col  = lane & 15;
  const int half = lane >> 4;

  v2f a;  a[0] = 1.f; a[1] = 1.f;          // all-ones A: layout-independent
  v8f c0 = {}, c1 = {}, c2 = {};

  const int groups = nblocks >> 2;          // nblocks is a multiple of 4
  for (int g = 0; g < groups; ++g) {
    const float* base = partials + (size_t)(g * 4) * PARTIAL_F;
    const int k0 = half;                    // slot->K map; any bijection OK
    const int k1 = 2 + half;
    v2f b0, b1, b2;
    b0[0] = base[k0 * PARTIAL_F + 0 * 16 + col];
    b0[1] = base[k1 * PARTIAL_F + 0 * 16 + col];
    b1[0] = base[k0 * PARTIAL_F + 1 * 16 + col];
    b1[1] = base[k1 * PARTIAL_F + 1 * 16 + col];
    b2[0] = base[k0 * PARTIAL_F + 2 * 16 + col];
    b2[1] = base[k1 * PARTIAL_F + 2 * 16 + col];
    c0 = __builtin_amdgcn_wmma_f32_16x16x4_f32(false, a, false, b0, (short)0, c0, false, false);
    c1 = __builtin_amdgcn_wmma_f32_16x16x4_f32(false, a, false, b1, (short)0, c1, false, false);
    c2 = __builtin_amdgcn_wmma_f32_16x16x4_f32(false, a, false, b2, (short)0, c2, false, false);
  }

  // D rows are all identical (A==ones); lanes 0..15 / VGPR 0 hold N=lane.
  const float cnt   = c0[0];
  const float accs  = c1[0];
  const float confs = c2[0];

  __shared__ float sg[16];
  if (lane < 16) {
    float gap = 0.f;
    if (lane < NBINS && cnt > 0.f) {
      float safe = fmaxf(cnt, 1.f);
      gap = fabsf(confs / safe - accs / safe) * (cnt / (float)n);
    }
    sg[lane] = gap;
  }
  __syncthreads();
  if (lane == 0) {
    float s = 0.f;
#pragma unroll
    for (int i = 0; i < 16; ++i) s += sg[i];
    out[0] = s;
  }
}

// ---------------------------------------------------------------------------
extern "C" void kernel_launch(void* const* d_in, const int* in_sizes, int n_in,
                              void* d_out, int out_size, void* d_ws, size_t ws_size,
                              hipStream_t stream) {
  const float* soft   = (const float*)d_in[0];
  const int*   labels = (const int*)d_in[1];
  float*       out    = (float*)d_out;
  const int    n      = in_sizes[1];        // number of rows (labels count)

  int nblocks = MAX_BLOCKS;                 // keep a multiple of 4
  while ((size_t)nblocks * PARTIAL_F * sizeof(float) > ws_size && nblocks > 4)
    nblocks >>= 1;

  float* partials = (float*)d_ws;
  ece_partial_kernel<<<nblocks, TPB, 0, stream>>>(soft, labels, partials, n, nblocks);
  ece_final_kernel<<<1, 32, 0, stream>>>(partials, out, nblocks, n);
}